// semantic_energy_loss_70085276336735
// MI455X (gfx1250) — compile-verified
//
#include <hip/hip_runtime.h>
#include <hip/hip_bf16.h>
#include <math.h>

#define B_    8
#define DIM_  70
#define KPAD  72
#define N_    1024
#define LAM1  2.0f
#define LAM2  0.05f
#define LAM3  0.2f
#define LAMP  0.2f

typedef __attribute__((ext_vector_type(2))) float v2f;
typedef __attribute__((ext_vector_type(8))) float v8f;

// workspace layout (float units)
// [0..8191]        invnorm        B*N
// [8192..16383]    sq             B*N
// [16384..24575]   labF (int)     B*N
// [24576..49151]   energy         B*3*N
// [49152..638975]  fpad           B*KPAD*N
// [638976..639007] scal: n1f[8] n2f[8] cnt[8] lam[8]
// [639008..639032] accum: fd[8] regsum[1] exclS[8] labpix[8]
static constexpr size_t OFF_INV  = 0;
static constexpr size_t OFF_SQ   = (size_t)B_ * N_;
static constexpr size_t OFF_LAB  = OFF_SQ + (size_t)B_ * N_;
static constexpr size_t OFF_EN   = OFF_LAB + (size_t)B_ * N_;
static constexpr size_t OFF_FP   = OFF_EN + (size_t)B_ * 3 * N_;
static constexpr size_t OFF_SCAL = OFF_FP + (size_t)B_ * KPAD * N_;
static constexpr size_t OFF_ACC  = OFF_SCAL + 32;

// ---------------------------------------------------------------- kernel 1
// Per-batch: counts, lab flags, norms, padded feature slab, prototype dots,
// the three energy maps (re, re1, re2), per-batch scalars, zero accumulators.
__global__ void k_stats(const float* __restrict__ feat, const int* __restrict__ labels,
                        const float* __restrict__ ptum, const float* __restrict__ ptis,
                        float* __restrict__ invnorm, float* __restrict__ sq,
                        int* __restrict__ labF, float* __restrict__ energy,
                        float* __restrict__ fpad, float* __restrict__ scal,
                        float* __restrict__ accum)
{
  const int b   = blockIdx.x;
  const int tid = threadIdx.x;
  __shared__ float s_pt[DIM_], s_ps[DIM_], s_s1[DIM_], s_s2[DIM_];
  __shared__ int s_n1, s_n2;
  const float* fb = feat + (size_t)b * DIM_ * N_;
  const int*   lb = labels + (size_t)b * N_;

  if (tid == 0) { s_n1 = 0; s_n2 = 0; }
  __syncthreads();

  int c1 = 0, c2 = 0;
  for (int n = tid; n < N_; n += blockDim.x) {
    int l = lb[n];
    c1 += (l == 1); c2 += (l == 2);
    labF[b * N_ + n] = (l == 1 || l == 2) ? 1 : 0;
  }
  atomicAdd(&s_n1, c1);
  atomicAdd(&s_n2, c2);
  if (tid < DIM_) { s_pt[tid] = ptum[tid]; s_ps[tid] = ptis[tid]; }
  __syncthreads();

  if (tid == 0) {
    float a = 0.f, c = 0.f;
    for (int d = 0; d < DIM_; ++d) { a += s_pt[d] * s_pt[d]; c += s_ps[d] * s_ps[d]; }
    a = rsqrtf(a); c = rsqrtf(c);
    for (int d = 0; d < DIM_; ++d) { s_pt[d] *= a; s_ps[d] *= c; }
  }
  for (int n = tid; n < N_; n += blockDim.x) {
    float s = 0.f;
    for (int d = 0; d < DIM_; ++d) { float v = fb[d * N_ + n]; s += v * v; }
    sq[b * N_ + n] = s;
    invnorm[b * N_ + n] = rsqrtf(s);
  }
  for (int idx = tid; idx < KPAD * N_; idx += blockDim.x) {
    int d = idx >> 10, n = idx & (N_ - 1);
    fpad[(size_t)b * KPAD * N_ + idx] = (d < DIM_) ? fb[d * N_ + n] : 0.f;
  }
  __syncthreads();

  for (int d = tid; d < DIM_; d += blockDim.x) {
    float a1 = 0.f, a2 = 0.f;
    for (int n = 0; n < N_; ++n) {
      int l = lb[n];
      if (l == 1 || l == 2) {
        float x = fb[d * N_ + n] * invnorm[b * N_ + n];
        if (l == 1) a1 += x; else a2 += x;
      }
    }
    s_s1[d] = a1; s_s2[d] = a2;
  }
  __syncthreads();

  const int n1c = s_n1, n2c = s_n2;
  const float n1f = n1c ? (float)n1c : 1e-10f;
  const float n2f = n2c ? (float)n2c : 1e-10f;
  for (int n = tid; n < N_; n += blockDim.x) {
    float t1 = 0.f, t2 = 0.f, tp = 0.f, ts = 0.f;
    for (int d = 0; d < DIM_; ++d) {
      float f = fb[d * N_ + n];
      t1 += f * s_s1[d]; t2 += f * s_s2[d]; tp += f * s_pt[d]; ts += f * s_ps[d];
    }
    const float inv = invnorm[b * N_ + n];
    const float xs1 = inv * t1, xs2 = inv * t2, ct = inv * tp, cs = inv * ts;
    const int lab = labF[b * N_ + n];
    // cosd@mask = 0.5*(count + xn·(Xn^T mask))
    float re1p = 0.5f * ((float)n1c + xs1) / n1f;
    float re1v = 1.f - (LAMP * ct + (1.f - LAMP) * ((n1c == 0) ? ct : re1p));
    float rev  = 1.f - 0.5f * ((float)(n1c + n2c) + xs1 + xs2) / (n1f + n2f);
    float re2p = 0.5f * ((float)n2c + xs2) / n2f;
    float re2v = 1.f - (LAMP * cs + (1.f - LAMP) * ((n2c == 0) ? cs : re2p));
    if (lab) { rev = 1.f; re1v = 1.f; re2v = 1.f; }
    energy[((size_t)b * 3 + 0) * N_ + n] = rev;
    energy[((size_t)b * 3 + 1) * N_ + n] = re1v;
    energy[((size_t)b * 3 + 2) * N_ + n] = re2v;
  }
  if (tid == 0) {
    scal[0 * B_ + b] = n1f;
    scal[1 * B_ + b] = n2f;
    int nL = n1c + n2c;
    scal[2 * B_ + b] = (float)nL * (float)(N_ - nL);      // pair count
    scal[3 * B_ + b] = n1f + n2f / ((float)N_ * 0.05f);   // lam_lab
    accum[b] = 0.f;        // fd
    accum[9 + b] = 0.f;    // exclS
    accum[17 + b] = 0.f;   // labpix
    if (b == 0) accum[8] = 0.f;  // regsum
  }
}

// ---------------------------------------------------------------- kernel 2
// Masked pairwise exp(3-d) sum via V_WMMA_F32_16X16X4_F32 Gram tiles.
// One wave per 16x16 output tile; K padded to 72 (18 WMMAs).
__global__ void k_pairwise(const float* __restrict__ fpad, const float* __restrict__ sq,
                           const int* __restrict__ labF, float* __restrict__ accum)
{
  const int tilesPerBatch = (N_ / 16) * (N_ / 16); // 4096
  const int gwave = blockIdx.x * (blockDim.x >> 5) + (threadIdx.x >> 5);
  const int lane  = threadIdx.x & 31;
  const int b = gwave / tilesPerBatch;
  const int t = gwave % tilesPerBatch;
  if (b >= B_) return;                       // wave-uniform
  const int i0 = (t >> 6) * 16, j0 = (t & 63) * 16;
  const int r = lane & 15;
  const int* lb = labF + b * N_;

  const int labRow = lb[i0 + r];
  const int labCol = lb[j0 + r];
  const bool any = (__ballot(labRow != 0) != 0ull) && (__ballot(labCol == 0) != 0ull);
  if (!any) return;                          // wave-uniform -> EXEC all-1s below

  const float* F = fpad + (size_t)b * KPAD * N_;
  const int kk = (lane >> 4) * 2;            // ISA A 16x4 f32 layout: K-pair by lane half
  v8f c = {};
  for (int k0 = 0; k0 < KPAD; k0 += 4) {
    v2f a, bm;
    a.x  = F[(size_t)(k0 + kk)     * N_ + i0 + r];
    a.y  = F[(size_t)(k0 + kk + 1) * N_ + i0 + r];
    bm.x = F[(size_t)(k0 + kk)     * N_ + j0 + r];
    bm.y = F[(size_t)(k0 + kk + 1) * N_ + j0 + r];
    c = __builtin_amdgcn_wmma_f32_16x16x4_f32(false, a, false, bm,
                                              (short)0, c, false, false);
  }

  float acc = 0.f;
  const int m = j0 + r;                      // C/D layout: N = lane%16
  const bool colU = (lb[m] == 0);
  const float sqm = sq[b * N_ + m];
  const int half = (lane >> 4) * 8;          // C/D layout: M = v + 8*(lane/16)
  if (colU) {
    for (int v = 0; v < 8; ++v) {
      const int row = i0 + half + v;
      if (lb[row]) {
        float d2 = sq[b * N_ + row] + sqm - 2.f * c[v];
        d2 = fmaxf(d2, 0.f);
        const float d = sqrtf(fmaxf(d2, 1e-12f));
        acc += expf(3.f - d);
      }
    }
  }
  for (int o = 16; o > 0; o >>= 1) acc += __shfl_xor(acc, o, 32);
  if (lane == 0) atomicAdd(&accum[b], acc);
}

// ---------------------------------------------------------------- kernel 3
// Streaming pass over [B,3,1024,1024]: bilinear-upsampled energy * CE + exclusion.
__global__ void k_pixels(const float* __restrict__ reg, const int* __restrict__ chm,
                         const float* __restrict__ energy, float* __restrict__ accum)
{
  const size_t gid = (size_t)blockIdx.x * blockDim.x + threadIdx.x;
  const int b   = (int)(gid >> 20);
  const int pix = (int)(gid & ((1u << 20) - 1));
  const int y = pix >> 10, x = pix & 1023;

  // half-pixel bilinear 32 -> 1024 with edge clamp
  const float sy = (y + 0.5f) * (1.f / 32.f) - 0.5f;
  const float sx = (x + 0.5f) * (1.f / 32.f) - 0.5f;
  const float fy = floorf(sy), fx = floorf(sx);
  const float wy = sy - fy, wx = sx - fx;
  int y0 = (int)fy, x0 = (int)fx;
  const int y1 = min(y0 + 1, 31), x1 = min(x0 + 1, 31);
  y0 = max(y0, 0); x0 = max(x0, 0);

  const float* eb = energy + (size_t)b * 3 * N_;
  const size_t base = (size_t)b * 3 * (1u << 20) + (size_t)y * 1024 + x;

  float rc[3], ec[3];
  int cv[3];
  for (int cch = 0; cch < 3; ++cch) {
    const float* e = eb + cch * N_;
    const float e00 = e[y0 * 32 + x0], e01 = e[y0 * 32 + x1];
    const float e10 = e[y1 * 32 + x0], e11 = e[y1 * 32 + x1];
    ec[cch] = (1.f - wy) * ((1.f - wx) * e00 + wx * e01)
            +        wy  * ((1.f - wx) * e10 + wx * e11);
    const size_t off = base + (size_t)cch * (1u << 20);
    rc[cch] = fminf(fmaxf(reg[off], 1e-10f), 1.0f);
    cv[cch] = chm[off];
  }
  const int h1 = cv[1] > 0, h2 = cv[2] > 0;
  const int labreg = (cv[1] + cv[2]) > 0;
  float heat[3] = { 1.f - (float)h1 - (float)h2, (float)h1, (float)h2 };

  float rl = 0.f, ex = 0.f;
  for (int cch = 0; cch < 3; ++cch) {
    rl += -(float)cv[cch] * logf(rc[cch]) * ec[cch];
    ex += (1.f - heat[cch]) * log1pf(rc[cch]);
  }
  ex *= (float)labreg;
  float lp = (float)labreg;

  for (int o = 16; o > 0; o >>= 1) {
    rl += __shfl_xor(rl, o, 32);
    ex += __shfl_xor(ex, o, 32);
    lp += __shfl_xor(lp, o, 32);
  }
  __shared__ float sred[8][3];
  const int w = threadIdx.x >> 5;
  if ((threadIdx.x & 31) == 0) { sred[w][0] = rl; sred[w][1] = ex; sred[w][2] = lp; }
  __syncthreads();
  if (threadIdx.x == 0) {
    float a = 0.f, bq = 0.f, cq = 0.f;
    for (int i = 0; i < 8; ++i) { a += sred[i][0]; bq += sred[i][1]; cq += sred[i][2]; }
    atomicAdd(&accum[8], a);
    atomicAdd(&accum[9 + b], bq);
    atomicAdd(&accum[17 + b], cq);
  }
}

// ---------------------------------------------------------------- kernel 4
__global__ void k_final(const float* __restrict__ scal, const float* __restrict__ accum,
                        float* __restrict__ out)
{
  const int lane = threadIdx.x;
  float str = 0.f, exc = 0.f;
  if (lane < B_) {
    const float cnt = scal[2 * B_ + lane];
    const float fd = accum[lane] / fmaxf(cnt, 1.f);
    str = scal[3 * B_ + lane] * fd;
    exc = accum[9 + lane] / accum[17 + lane] + 1e-10f;
  }
  for (int o = 16; o > 0; o >>= 1) { str += __shfl_xor(str, o, 32); exc += __shfl_xor(exc, o, 32); }
  if (lane == 0) {
    const float stretched = str / (float)B_;
    const float exclusion = exc / (float)B_;
    const float regloss = accum[8] / (float)((size_t)B_ * (1u << 20));
    out[0] = LAM1 * regloss + LAM2 * stretched + LAM3 * exclusion;
    out[1] = regloss;
    out[2] = stretched;
    out[3] = exclusion;
  }
}

extern "C" void kernel_launch(void* const* d_in, const int* in_sizes, int n_in,
                              void* d_out, int out_size, void* d_ws, size_t ws_size,
                              hipStream_t stream) {
  (void)in_sizes; (void)n_in; (void)out_size; (void)ws_size;
  const float* feat   = (const float*)d_in[0];  // [8,70,32,32]
  const int*   labels = (const int*)d_in[1];    // [8,1024]
  const float* reg    = (const float*)d_in[2];  // [8,3,1024,1024]
  const int*   chm    = (const int*)d_in[3];    // [8,3,1024,1024]
  const float* ptum   = (const float*)d_in[4];  // [70]
  const float* ptis   = (const float*)d_in[5];  // [70]
  float* out = (float*)d_out;

  float* w       = (float*)d_ws;
  float* invnorm = w + OFF_INV;
  float* sq      = w + OFF_SQ;
  int*   labF    = (int*)(w + OFF_LAB);
  float* energy  = w + OFF_EN;
  float* fpad    = w + OFF_FP;
  float* scal    = w + OFF_SCAL;
  float* accum   = w + OFF_ACC;

  k_stats<<<B_, 256, 0, stream>>>(feat, labels, ptum, ptis,
                                  invnorm, sq, labF, energy, fpad, scal, accum);
  // B_ * 4096 tiles, 8 waves (tiles) per 256-thread block
  k_pairwise<<<(B_ * 4096) / 8, 256, 0, stream>>>(fpad, sq, labF, accum);
  // 8 * 2^20 pixels, 1 per thread
  k_pixels<<<(B_ * (1 << 20)) / 256, 256, 0, stream>>>(reg, chm, energy, accum);
  k_final<<<1, 32, 0, stream>>>(scal, accum, out);
}